// PScanTriton_35768487641424
// MI455X (gfx1250) — compile-verified
//
#include <hip/hip_runtime.h>

// y_t = a_t * y_{t-1} + x_t  (complex), B=4, L=2048, C=1024, fp32.
// Bandwidth-bound (192 MiB min traffic, ~0.33 FLOP/byte) -> 3-pass blocked
// scan, streaming loads staged through LDS with gfx1250 async-to-LDS loads.
// Pass-3 re-reads of A/X are expected to hit the 192 MB L2 (A+X = 128 MiB);
// pass-3 output stores are non-temporal so they don't evict A/X lines.

#define B_      4
#define L_      2048
#define C_      1024
#define T_      32              // tiles along time axis
#define TILE_   (L_ / T_)       // 64 steps per tile
#define DEPTH_  8               // async pipeline depth (steps in flight)
#define BLOCK_  256

// ---------- gfx1250 async global->LDS helpers ----------

__device__ __forceinline__ void issue_async_pair(uint32_t lds_a, uint32_t lds_x,
                                                 uint32_t goff,
                                                 unsigned long long abase,
                                                 unsigned long long xbase) {
  // Two 8-byte async loads (A pair, X pair) into this lane's LDS slot.
  // GVS mode: mem_addr = SADDR(64b) + VADDR(32b offset).
  asm volatile(
      "global_load_async_to_lds_b64 %0, %2, %3\n\t"
      "global_load_async_to_lds_b64 %1, %2, %4"
      :
      : "v"(lds_a), "v"(lds_x), "v"(goff), "s"(abase), "s"(xbase)
      : "memory");
}

__device__ __forceinline__ void wait_asynccnt_14() {
  asm volatile("s_wait_asynccnt 14" ::: "memory");
}
__device__ __forceinline__ void wait_asynccnt_0() {
  asm volatile("s_wait_asynccnt 0" ::: "memory");
}

// Force v's ds_load results to be materialized (s_wait_dscnt) *before* the
// next async issue overwrites the slot.
__device__ __forceinline__ void consume_fence(const float4& v) {
  asm volatile("" :: "v"(v.x), "v"(v.y), "v"(v.z), "v"(v.w));
}

// Non-temporal 8-byte store (output is write-once, never re-read on device).
__device__ __forceinline__ void store_nt_f2(float2* p, float r, float i) {
  union { float f[2]; unsigned long long u; } pk;
  pk.f[0] = r; pk.f[1] = i;
  __builtin_nontemporal_store(pk.u, (unsigned long long*)p);
}

// ---------- Pass 1: per-tile aggregates (P = prod a, Y = local scan end) ----

__global__ __launch_bounds__(BLOCK_) void pscan_pass1(
    const float* __restrict__ A, const float* __restrict__ X,
    float4* __restrict__ agg) {
  __shared__ float4 smem[DEPTH_ * BLOCK_];

  const uint32_t tid  = threadIdx.x;
  const uint32_t g    = blockIdx.x * BLOCK_ + tid;
  const uint32_t c    = g & (C_ - 1);
  const uint32_t tile = (g >> 10) & (T_ - 1);
  const uint32_t b    = g >> 15;
  const uint32_t t0   = tile * TILE_;

  const unsigned long long abase = (unsigned long long)A;
  const unsigned long long xbase = (unsigned long long)X;
  const uint32_t lbase = (uint32_t)(uintptr_t)(void*)smem;
  const uint32_t gstep = C_ * 8u;                      // bytes per time step
  uint32_t off = ((b * L_ + t0) * C_ + c) * 8u;        // byte offset (A and X)

  // prologue: fill the pipeline
  for (int d = 0; d < DEPTH_; ++d) {
    uint32_t la = lbase + ((uint32_t)d * BLOCK_ + tid) * 16u;
    issue_async_pair(la, la + 8u, off, abase, xbase);
    off += gstep;
  }

  float Pr = 1.f, Pi = 0.f, Yr = 0.f, Yi = 0.f;

  for (int i = 0; i < TILE_ - DEPTH_; ++i) {
    wait_asynccnt_14();                                // oldest step landed
    const uint32_t slot = (uint32_t)(i & (DEPTH_ - 1));
    float4 v = smem[slot * BLOCK_ + tid];
    consume_fence(v);
    uint32_t la = lbase + (slot * BLOCK_ + tid) * 16u;
    issue_async_pair(la, la + 8u, off, abase, xbase);  // step i+DEPTH_
    off += gstep;
    float nPr = v.x * Pr - v.y * Pi;
    float nPi = v.x * Pi + v.y * Pr;
    float nYr = v.x * Yr - v.y * Yi + v.z;
    float nYi = v.x * Yi + v.y * Yr + v.w;
    Pr = nPr; Pi = nPi; Yr = nYr; Yi = nYi;
  }
  wait_asynccnt_0();
  for (int i = TILE_ - DEPTH_; i < TILE_; ++i) {
    float4 v = smem[(uint32_t)(i & (DEPTH_ - 1)) * BLOCK_ + tid];
    float nPr = v.x * Pr - v.y * Pi;
    float nPi = v.x * Pi + v.y * Pr;
    float nYr = v.x * Yr - v.y * Yi + v.z;
    float nYi = v.x * Yi + v.y * Yr + v.w;
    Pr = nPr; Pi = nPi; Yr = nYr; Yi = nYi;
  }

  agg[(b * T_ + tile) * C_ + c] = make_float4(Pr, Pi, Yr, Yi);
}

// ---------- Pass 2: exclusive scan over tile aggregates per (b,c) ----------

__global__ __launch_bounds__(256) void pscan_pass2(
    const float4* __restrict__ agg, float2* __restrict__ carry) {
  const uint32_t g = blockIdx.x * 256u + threadIdx.x;  // 0 .. B_*C_-1
  const uint32_t c = g & (C_ - 1);
  const uint32_t b = g >> 10;

  float Pr = 1.f, Pi = 0.f, Yr = 0.f, Yi = 0.f;
  for (int k = 0; k < T_; ++k) {
    const uint32_t idx = (b * T_ + k) * C_ + c;
    carry[idx] = make_float2(Yr, Yi);                  // y entering tile k
    float4 v = agg[idx];                               // (Pk, Yk)
    float nPr = v.x * Pr - v.y * Pi;                   // P' = Pk * P
    float nPi = v.x * Pi + v.y * Pr;
    float nYr = v.x * Yr - v.y * Yi + v.z;             // Y' = Pk*Y + Yk
    float nYi = v.x * Yi + v.y * Yr + v.w;
    Pr = nPr; Pi = nPi; Yr = nYr; Yi = nYi;
  }
}

// ---------- Pass 3: replay recurrence with carry, write output -------------

__global__ __launch_bounds__(BLOCK_) void pscan_pass3(
    const float* __restrict__ A, const float* __restrict__ X,
    const float2* __restrict__ carry, float2* __restrict__ out) {
  __shared__ float4 smem[DEPTH_ * BLOCK_];

  const uint32_t tid  = threadIdx.x;
  const uint32_t g    = blockIdx.x * BLOCK_ + tid;
  const uint32_t c    = g & (C_ - 1);
  const uint32_t tile = (g >> 10) & (T_ - 1);
  const uint32_t b    = g >> 15;
  const uint32_t t0   = tile * TILE_;

  const unsigned long long abase = (unsigned long long)A;
  const unsigned long long xbase = (unsigned long long)X;
  const uint32_t lbase = (uint32_t)(uintptr_t)(void*)smem;
  const uint32_t gstep = C_ * 8u;
  uint32_t off = ((b * L_ + t0) * C_ + c) * 8u;

  for (int d = 0; d < DEPTH_; ++d) {
    uint32_t la = lbase + ((uint32_t)d * BLOCK_ + tid) * 16u;
    issue_async_pair(la, la + 8u, off, abase, xbase);
    off += gstep;
  }

  const float2 yc = carry[(b * T_ + tile) * C_ + c];
  float yr = yc.x, yi = yc.y;
  uint32_t oidx = (b * L_ + t0) * C_ + c;              // float2 index

  for (int i = 0; i < TILE_ - DEPTH_; ++i) {
    wait_asynccnt_14();
    const uint32_t slot = (uint32_t)(i & (DEPTH_ - 1));
    float4 v = smem[slot * BLOCK_ + tid];
    consume_fence(v);
    uint32_t la = lbase + (slot * BLOCK_ + tid) * 16u;
    issue_async_pair(la, la + 8u, off, abase, xbase);
    off += gstep;
    float nyr = v.x * yr - v.y * yi + v.z;
    float nyi = v.x * yi + v.y * yr + v.w;
    yr = nyr; yi = nyi;
    store_nt_f2(out + oidx, yr, yi);
    oidx += C_;
  }
  wait_asynccnt_0();
  for (int i = TILE_ - DEPTH_; i < TILE_; ++i) {
    float4 v = smem[(uint32_t)(i & (DEPTH_ - 1)) * BLOCK_ + tid];
    float nyr = v.x * yr - v.y * yi + v.z;
    float nyi = v.x * yi + v.y * yr + v.w;
    yr = nyr; yi = nyi;
    store_nt_f2(out + oidx, yr, yi);
    oidx += C_;
  }
}

// ---------- launch ----------

extern "C" void kernel_launch(void* const* d_in, const int* in_sizes, int n_in,
                              void* d_out, int out_size, void* d_ws, size_t ws_size,
                              hipStream_t stream) {
  (void)in_sizes; (void)n_in; (void)out_size; (void)ws_size;
  const float* A = (const float*)d_in[0];
  const float* X = (const float*)d_in[1];

  float4* agg   = (float4*)d_ws;                                   // 2 MiB
  float2* carry = (float2*)((char*)d_ws +
                            (size_t)B_ * T_ * C_ * sizeof(float4)); // +1 MiB

  const int tasks = B_ * T_ * C_;          // 131072 lane-tasks
  pscan_pass1<<<tasks / BLOCK_, BLOCK_, 0, stream>>>(A, X, agg);
  pscan_pass2<<<(B_ * C_) / 256, 256, 0, stream>>>(agg, carry);
  pscan_pass3<<<tasks / BLOCK_, BLOCK_, 0, stream>>>(A, X, carry,
                                                     (float2*)d_out);
}